// PointPillarVoxelization_22273700397338
// MI455X (gfx1250) — compile-verified
//
#include <hip/hip_runtime.h>
#include <stdint.h>

// ---- problem constants (match reference) ----
#define VSX 0.16f
#define VSY 0.16f
#define VSZ 4.0f
#define RMINX 0.0f
#define RMINY -39.68f
#define RMINZ -3.0f
#define NXV 432
#define NYV 496
#define NZV 1
#define BIGV (NXV * NYV * NZV)      // 214272 possible pillars
#define MAXV 12000                  // MAX_VOXELS
#define MAXP 100                    // MAX_NUM_POINTS
#define CAPV 128                    // per-pillar capture capacity (>= max realistic count)
#define SCAN_CHUNK 2048
#define NBLK ((BIGV + SCAN_CHUNK - 1) / SCAN_CHUNK)   // 105

// ---- gfx1250 async global->LDS copy + split wait counters ----
typedef int v4i __attribute__((vector_size(16)));
typedef __attribute__((address_space(1))) v4i gas_v4i;   // global
typedef __attribute__((address_space(3))) v4i las_v4i;   // LDS

#if defined(__has_builtin) && __has_builtin(__builtin_amdgcn_global_load_async_to_lds_b128)
#define ASYNC_LOAD_B128(gptr, lptr)                                        \
  __builtin_amdgcn_global_load_async_to_lds_b128(                          \
      (gas_v4i*)(void*)(gptr), (las_v4i*)(void*)(lptr), 0, 0)
#else
#define ASYNC_LOAD_B128(gptr, lptr)                                        \
  asm volatile("global_load_async_to_lds_b128 %0, %1, off"                 \
               :: "v"((las_v4i*)(void*)(lptr)),                            \
                  "v"((const void*)(gptr))                                 \
               : "memory")
#endif

#if defined(__has_builtin) && __has_builtin(__builtin_amdgcn_s_wait_asynccnt)
#define WAIT_ASYNCCNT(N) __builtin_amdgcn_s_wait_asynccnt(N)
#else
#define WAIT_ASYNCCNT(N) asm volatile("s_wait_asynccnt %0" :: "n"(N) : "memory")
#endif

// ---------------- kernel 0: zero outputs + counters ----------------
__global__ void k_init(float* out, int out_elems, int* counts, int* cursor) {
    long long stride = (long long)gridDim.x * blockDim.x;
    for (long long i = (long long)blockIdx.x * blockDim.x + threadIdx.x;
         i < out_elems || i < BIGV; i += stride) {
        if (i < out_elems) out[i] = 0.0f;
        if (i < BIGV)      counts[i] = 0;
        if (i < MAXV)      cursor[i] = 0;
    }
}

// ---------------- kernel 1: per-key counts (async-LDS staged stream) ----------------
__global__ void k_count(const float4* __restrict__ pts, int n, int* __restrict__ counts) {
    __shared__ float4 buf[2][256];
    const int t = threadIdx.x;
    const long long stride = (long long)gridDim.x * blockDim.x;
    long long i0 = (long long)blockIdx.x * blockDim.x + t;
    int phase = 0;
    if (i0 < n) ASYNC_LOAD_B128(pts + i0, &buf[0][t]);
    for (long long i = i0; i < n; i += stride) {
        const long long nx = i + stride;
        const bool has_next = (nx < n);
        if (has_next) ASYNC_LOAD_B128(pts + nx, &buf[phase ^ 1][t]);
        if (has_next) { WAIT_ASYNCCNT(1); } else { WAIT_ASYNCCNT(0); }
        asm volatile("" ::: "memory");
        float4 p = buf[phase][t];
        phase ^= 1;
        int cx = (int)floorf((p.x - RMINX) / VSX);
        int cy = (int)floorf((p.y - RMINY) / VSY);
        int cz = (int)floorf((p.z - RMINZ) / VSZ);
        if (cx >= 0 && cx < NXV && cy >= 0 && cy < NYV && cz >= 0 && cz < NZV) {
            int key = (cz * NYV + cy) * NXV + cx;
            atomicAdd(&counts[key], 1);
        }
    }
}

// ---------------- kernel 2: per-block occupancy sums ----------------
__global__ void k_scan_blocks(const int* __restrict__ counts, int* __restrict__ bsums) {
    __shared__ int red[256];
    const int b = blockIdx.x, t = threadIdx.x;
    const int base = b * SCAN_CHUNK + t * 8;
    int s = 0;
    for (int j = 0; j < 8; ++j) {
        int k = base + j;
        if (k < BIGV) s += (counts[k] > 0);
    }
    red[t] = s;
    __syncthreads();
    for (int off = 128; off > 0; off >>= 1) {
        if (t < off) red[t] += red[t + off];
        __syncthreads();
    }
    if (t == 0) bsums[b] = red[0];
}

// ---------------- kernel 3: exclusive scan of block sums (tiny) ----------------
__global__ void k_scan_sums(int* bsums, int nblk) {
    if (blockIdx.x == 0 && threadIdx.x == 0) {
        int acc = 0;
        for (int i = 0; i < nblk; ++i) { int v = bsums[i]; bsums[i] = acc; acc += v; }
    }
}

// ---------------- kernel 4: assign slots, emit coords + num_points ----------------
__global__ void k_assign(const int* __restrict__ counts, const int* __restrict__ bbase,
                         int* __restrict__ vslot,
                         float* __restrict__ out_coords, float* __restrict__ out_nump) {
    __shared__ int sc[256];
    const int b = blockIdx.x, t = threadIdx.x;
    const int base = b * SCAN_CHUNK + t * 8;
    int local[8];
    int s = 0;
    for (int j = 0; j < 8; ++j) {
        int k = base + j;
        int occ = (k < BIGV) ? (counts[k] > 0) : 0;
        local[j] = s;
        s += occ;
    }
    sc[t] = s;
    __syncthreads();
    for (int off = 1; off < 256; off <<= 1) {   // Hillis-Steele inclusive scan
        int v = (t >= off) ? sc[t - off] : 0;
        __syncthreads();
        sc[t] += v;
        __syncthreads();
    }
    const int texcl = (t == 0) ? 0 : sc[t - 1];
    const int start = bbase[b] + texcl;
    for (int j = 0; j < 8; ++j) {
        int k = base + j;
        if (k >= BIGV) break;
        int c = counts[k];
        int slot = -1;
        if (c > 0) {
            int cand = start + local[j];
            if (cand < MAXV) {
                slot = cand;
                int cx = k % NXV;
                int cy = (k / NXV) % NYV;
                int cz = k / (NXV * NYV);
                out_coords[slot * 3 + 0] = (float)cz;
                out_coords[slot * 3 + 1] = (float)cy;
                out_coords[slot * 3 + 2] = (float)cx;
                out_nump[slot] = (float)c;
            }
        }
        vslot[k] = slot;
    }
}

// ---------------- kernel 5: scatter point indices into per-slot lists ----------------
__global__ void k_scatter(const float4* __restrict__ pts, int n,
                          const int* __restrict__ vslot,
                          int* __restrict__ cursor, int* __restrict__ lists) {
    long long stride = (long long)gridDim.x * blockDim.x;
    for (long long i = (long long)blockIdx.x * blockDim.x + threadIdx.x; i < n; i += stride) {
        float4 p = pts[i];
        int cx = (int)floorf((p.x - RMINX) / VSX);
        int cy = (int)floorf((p.y - RMINY) / VSY);
        int cz = (int)floorf((p.z - RMINZ) / VSZ);
        if (cx >= 0 && cx < NXV && cy >= 0 && cy < NYV && cz >= 0 && cz < NZV) {
            int key = (cz * NYV + cy) * NXV + cx;
            int s = vslot[key];
            if (s >= 0) {
                int pos = atomicAdd(&cursor[s], 1);
                if (pos < CAPV) lists[s * CAPV + pos] = (int)i;
            }
        }
    }
}

// ---------------- kernel 6: deterministic rank by index-sort, emit features ----------------
__global__ void k_emit(const float4* __restrict__ pts,
                       const int* __restrict__ cursor, const int* __restrict__ lists,
                       float* __restrict__ out_voxels) {
    __shared__ int sidx[CAPV];
    const int s = blockIdx.x;
    const int t = threadIdx.x;             // blockDim.x == CAPV (128 = 4 wave32)
    const int cnt = cursor[s];
    const int m = cnt < CAPV ? cnt : CAPV;
    if (t < m) sidx[t] = lists[s * CAPV + t];
    __syncthreads();
    if (t < m) {
        const int my = sidx[t];
        int rank = 0;
        for (int j = 0; j < m; ++j) rank += (sidx[j] < my);   // indices are distinct
        if (rank < MAXP) {
            float4 p = pts[my];
            float4* dst = (float4*)(out_voxels + ((size_t)s * MAXP + rank) * 4);
            *dst = p;
        }
    }
}

extern "C" void kernel_launch(void* const* d_in, const int* in_sizes, int n_in,
                              void* d_out, int out_size, void* d_ws, size_t ws_size,
                              hipStream_t stream) {
    (void)n_in; (void)ws_size;
    const float4* pts = (const float4*)d_in[0];
    const int n = in_sizes[0] / 4;

    float* out        = (float*)d_out;
    float* out_voxels = out;                                   // [12000,100,4]
    float* out_coords = out + (size_t)MAXV * MAXP * 4;         // [12000,3]
    float* out_nump   = out_coords + (size_t)MAXV * 3;         // [12000]

    char* w = (char*)d_ws;
    int* counts = (int*)w;  w += sizeof(int) * (size_t)BIGV;
    int* vslot  = (int*)w;  w += sizeof(int) * (size_t)BIGV;
    int* bsums  = (int*)w;  w += sizeof(int) * (size_t)(NBLK + 1);
    w = (char*)(((uintptr_t)w + 15) & ~(uintptr_t)15);
    int* cursor = (int*)w;  w += sizeof(int) * (size_t)MAXV;
    w = (char*)(((uintptr_t)w + 15) & ~(uintptr_t)15);
    int* lists  = (int*)w;  // MAXV * CAPV ints (~6.1 MB); total ws use ~7.9 MB

    k_init       <<<2048, 256, 0, stream>>>(out, out_size, counts, cursor);
    k_count      <<<2048, 256, 0, stream>>>(pts, n, counts);
    k_scan_blocks<<<NBLK, 256, 0, stream>>>(counts, bsums);
    k_scan_sums  <<<1,    1,   0, stream>>>(bsums, NBLK);
    k_assign     <<<NBLK, 256, 0, stream>>>(counts, bsums, vslot, out_coords, out_nump);
    k_scatter    <<<2048, 256, 0, stream>>>(pts, n, vslot, cursor, lists);
    k_emit       <<<MAXV, CAPV, 0, stream>>>(pts, cursor, lists, out_voxels);
}